// TaskAlignedAssigner_48352741818732
// MI455X (gfx1250) — compile-verified
//
#include <hip/hip_runtime.h>
#include <hip/hip_bf16.h>
#include <stdint.h>

#define NBINS   1024
#define TOPKK   10
#define G_MAX   64
#define CAP     2048
#define BLK     256

// ---- CDNA5 async global->LDS staging (guarded; falls back to plain LDS stores)
#if defined(__HIP_DEVICE_COMPILE__) && __has_builtin(__builtin_amdgcn_global_load_async_to_lds_b128)
#define USE_ASYNC_LDS 1
#else
#define USE_ASYNC_LDS 0
#endif

#if USE_ASYNC_LDS
typedef int v4i_async __attribute__((__vector_size__(4 * sizeof(int))));
typedef __attribute__((address_space(1))) v4i_async* as1_v4i_ptr;
typedef __attribute__((address_space(3))) v4i_async* as3_v4i_ptr;
#endif

__device__ __forceinline__ void wait_async_lds() {
#if USE_ASYNC_LDS
#if __has_builtin(__builtin_amdgcn_s_wait_asynccnt)
    __builtin_amdgcn_s_wait_asynccnt(0);
#else
    asm volatile("s_wait_asynccnt 0" ::: "memory");
#endif
#endif
}

// One 16-byte async global->LDS copy per calling lane (fallback: register copy).
__device__ __forceinline__ void copy16_g2l(const void* gsrc, void* ldst) {
#if USE_ASYNC_LDS
    __builtin_amdgcn_global_load_async_to_lds_b128(
        (as1_v4i_ptr)(uintptr_t)gsrc, (as3_v4i_ptr)(uint32_t)(uintptr_t)ldst, 0, 0);
#else
    *(uint4*)ldst = *(const uint4*)gsrc;
#endif
}

// Stage gt boxes (G*16B) into LDS via async DMA, derive areas; stage labels.
__device__ __forceinline__ void stage_gts(const float4* __restrict__ gt_bboxes,
                                          const int* __restrict__ gt_labels,
                                          int G, float4* sBox, float* sAreaG, int* sLbl)
{
    const int tid = threadIdx.x;
    if (tid < G) copy16_g2l(gt_bboxes + tid, sBox + tid);
    wait_async_lds();
    if (tid < G) sLbl[tid] = gt_labels[tid];
    __syncthreads();
    if (tid < G) {
        float4 b = sBox[tid];
        sAreaG[tid] = (b.z - b.x) * (b.w - b.y);
    }
    __syncthreads();
}

__device__ __forceinline__ float metric_val(float ax, float ay,
                                            float px0, float py0, float px1, float py1,
                                            float areaP, float cls,
                                            float4 gb, float areaG)
{
    bool in_gt = (ax >= gb.x) && (ax <= gb.z) && (ay >= gb.y) && (ay <= gb.w);
    float ltx = fmaxf(px0, gb.x), lty = fmaxf(py0, gb.y);
    float rbx = fminf(px1, gb.z), rby = fminf(py1, gb.w);
    float w = fmaxf(rbx - ltx, 0.0f), h = fmaxf(rby - lty, 0.0f);
    float inter = w * h;
    float iou = inter / (areaP + areaG - inter);
    float i2 = iou * iou;
    float i6 = i2 * i2 * i2;          // BETA = 6
    return (in_gt ? cls : 0.0f) * i6; // ALPHA = 1
}

__device__ __forceinline__ uint32_t metric_bin(float m) {
    uint32_t b = __float_as_uint(m) >> 20;  // positive floats: bits are monotonic
    return b > (NBINS - 1) ? (NBINS - 1) : b;
}

// ---------------- kernel Z: zero the counter region of the workspace ----------------
__global__ void k_zero(uint32_t* __restrict__ p, int n) {
    int i = blockIdx.x * blockDim.x + threadIdx.x;
    if (i < n) p[i] = 0u;
}

// ---------------- kernel A: per-gt histogram of positive metrics ----------------
__global__ void __launch_bounds__(BLK)
k_hist(const float* __restrict__ pd_scores, const float4* __restrict__ pd_bboxes,
       const float2* __restrict__ anc, const float4* __restrict__ gt_bboxes,
       const int* __restrict__ gt_labels, uint32_t* __restrict__ hist,
       int A, int G, int C)
{
    __shared__ float4 sBox[G_MAX];
    __shared__ float  sAreaG[G_MAX];
    __shared__ int    sLbl[G_MAX];
    stage_gts(gt_bboxes, gt_labels, G, sBox, sAreaG, sLbl);

    int a = blockIdx.x * blockDim.x + threadIdx.x;
    if (a >= A) return;
    float2 p  = anc[a];
    float4 pb = pd_bboxes[a];
    float areaP = (pb.z - pb.x) * (pb.w - pb.y);
    float cls0  = pd_scores[(size_t)a * C];

    for (int g = 0; g < G; ++g) {
        float cls = (C == 1) ? cls0 : pd_scores[(size_t)a * C + sLbl[g]];
        float m = metric_val(p.x, p.y, pb.x, pb.y, pb.z, pb.w, areaP, cls, sBox[g], sAreaG[g]);
        if (m > 0.0f)
            atomicAdd((unsigned int*)&hist[g * NBINS + metric_bin(m)], 1u);
    }
}

// ---------------- kernel B: locate boundary bin + residual rank per gt ----------------
// One block per gt: async-stage the 4KB histogram row to LDS, build 4-bin chunk
// sums, then a short LDS-latency scan instead of ~1024 dependent global loads.
__global__ void __launch_bounds__(BLK)
k_findbin(const uint32_t* __restrict__ hist,
          uint32_t* __restrict__ binIdx, uint32_t* __restrict__ needArr, int G)
{
    __shared__ uint32_t sBins[NBINS];
    __shared__ uint32_t sChunk[BLK];
    const int g = blockIdx.x;
    if (g >= G) return;
    const int t = threadIdx.x;

    copy16_g2l(hist + (size_t)g * NBINS + 4 * t, &sBins[4 * t]);
    wait_async_lds();
    __syncthreads();

    sChunk[t] = sBins[4 * t] + sBins[4 * t + 1] + sBins[4 * t + 2] + sBins[4 * t + 3];
    __syncthreads();

    if (t == 0) {
        uint32_t cum = 0, bsel = 0xFFFFFFFFu, need = 0;
        for (int cidx = BLK - 1; cidx >= 0; --cidx) {
            uint32_t cs = sChunk[cidx];
            if (cum + cs >= TOPKK) {
                for (int b = 4 * cidx + 3; b >= 4 * cidx; --b) {
                    uint32_t c = sBins[b];
                    if (cum + c >= TOPKK) { bsel = (uint32_t)b; need = TOPKK - cum; break; }
                    cum += c;
                }
                break;  // bsel found within this chunk (guaranteed)
            }
            cum += cs;
        }
        binIdx[g] = bsel;   // 0xFFFFFFFF => fewer than TOPK positives: accept all positives
        needArr[g] = need;
    }
}

// ---------------- kernel C: collect boundary-bin values ----------------
__global__ void __launch_bounds__(BLK)
k_collect(const float* __restrict__ pd_scores, const float4* __restrict__ pd_bboxes,
          const float2* __restrict__ anc, const float4* __restrict__ gt_bboxes,
          const int* __restrict__ gt_labels, const uint32_t* __restrict__ binIdx,
          uint32_t* __restrict__ candCount, uint32_t* __restrict__ cand,
          int A, int G, int C)
{
    __shared__ float4 sBox[G_MAX];
    __shared__ float  sAreaG[G_MAX];
    __shared__ int    sLbl[G_MAX];
    __shared__ uint32_t sBin[G_MAX];
    if (threadIdx.x < G) sBin[threadIdx.x] = binIdx[threadIdx.x];
    stage_gts(gt_bboxes, gt_labels, G, sBox, sAreaG, sLbl);

    int a = blockIdx.x * blockDim.x + threadIdx.x;
    if (a >= A) return;
    float2 p  = anc[a];
    float4 pb = pd_bboxes[a];
    float areaP = (pb.z - pb.x) * (pb.w - pb.y);
    float cls0  = pd_scores[(size_t)a * C];

    for (int g = 0; g < G; ++g) {
        float cls = (C == 1) ? cls0 : pd_scores[(size_t)a * C + sLbl[g]];
        float m = metric_val(p.x, p.y, pb.x, pb.y, pb.z, pb.w, areaP, cls, sBox[g], sAreaG[g]);
        if (m > 0.0f && metric_bin(m) == sBin[g]) {
            uint32_t pos = atomicAdd((unsigned int*)&candCount[g], 1u);
            if (pos < CAP) cand[(size_t)g * CAP + pos] = __float_as_uint(m);
        }
    }
}

// ---------------- kernel D: exact r-th largest inside boundary bin ----------------
__global__ void k_thresh(const uint32_t* __restrict__ binIdx, const uint32_t* __restrict__ needArr,
                         const uint32_t* __restrict__ candCount, const uint32_t* __restrict__ cand,
                         uint32_t* __restrict__ thrBits, int G)
{
    int g = blockIdx.x * blockDim.x + threadIdx.x;
    if (g >= G) return;
    if (binIdx[g] == 0xFFFFFFFFu) { thrBits[g] = 0u; return; }
    uint32_t c = candCount[g]; if (c > CAP) c = CAP;
    uint32_t r = needArr[g];   // 1..TOPK
    uint32_t top[TOPKK];
#pragma unroll
    for (int i = 0; i < TOPKK; ++i) top[i] = 0u;
    const uint32_t* cv = cand + (size_t)g * CAP;
    for (uint32_t i = 0; i < c; ++i) {
        uint32_t v = cv[i];
        if (v > top[TOPKK - 1]) {
            int j = TOPKK - 1;
            while (j > 0 && top[j - 1] < v) { top[j] = top[j - 1]; --j; }
            top[j] = v;
        }
    }
    thrBits[g] = top[r - 1];
}

// ---------------- kernel E: final fg_mask + assigned_gt_idx ----------------
__global__ void __launch_bounds__(BLK)
k_final(const float* __restrict__ pd_scores, const float4* __restrict__ pd_bboxes,
        const float2* __restrict__ anc, const float4* __restrict__ gt_bboxes,
        const int* __restrict__ gt_labels, const uint32_t* __restrict__ thrBits,
        float* __restrict__ out, int A, int G, int C)
{
    __shared__ float4 sBox[G_MAX];
    __shared__ float  sAreaG[G_MAX];
    __shared__ int    sLbl[G_MAX];
    __shared__ uint32_t sThr[G_MAX];
    if (threadIdx.x < G) sThr[threadIdx.x] = thrBits[threadIdx.x];
    stage_gts(gt_bboxes, gt_labels, G, sBox, sAreaG, sLbl);

    int a = blockIdx.x * blockDim.x + threadIdx.x;
    if (a >= A) return;
    float2 p  = anc[a];
    float4 pb = pd_bboxes[a];
    float areaP = (pb.z - pb.x) * (pb.w - pb.y);
    float cls0  = pd_scores[(size_t)a * C];

    float best = -1.0f;
    int bestg = 0;
    bool fg = false;
    for (int g = 0; g < G; ++g) {
        float cls = (C == 1) ? cls0 : pd_scores[(size_t)a * C + sLbl[g]];
        float m = metric_val(p.x, p.y, pb.x, pb.y, pb.z, pb.w, areaP, cls, sBox[g], sAreaG[g]);
        if (m > best) { best = m; bestg = g; }              // first-max tie rule (argmax)
        fg = fg || (m > 0.0f && __float_as_uint(m) >= sThr[g]);
    }
    out[a]     = fg ? 1.0f : 0.0f;           // fg_mask
    out[A + a] = fg ? (float)bestg : 0.0f;   // assigned_gt_idx (0 when !fg)
}

extern "C" void kernel_launch(void* const* d_in, const int* in_sizes, int n_in,
                              void* d_out, int out_size, void* d_ws, size_t ws_size,
                              hipStream_t stream) {
    const float* pd_scores = (const float*)d_in[0];
    const float* pd_bboxes = (const float*)d_in[1];
    const float* anc       = (const float*)d_in[2];
    const int*   gt_labels = (const int*)d_in[3];
    const float* gt_bboxes = (const float*)d_in[4];

    const int A = in_sizes[2] / 2;           // anc_points is (A,2)
    const int G = in_sizes[3];               // gt_labels is (G,)
    const int C = in_sizes[0] / A;           // pd_scores is (A,C) ; C == 1 here

    // workspace layout (uint32): hist[G*NBINS] | binIdx[G] | need[G] | thr[G] | cnt[G] | cand[G*CAP]
    uint32_t* hist    = (uint32_t*)d_ws;
    uint32_t* binIdx  = hist + (size_t)G * NBINS;
    uint32_t* needArr = binIdx + G;
    uint32_t* thr     = needArr + G;
    uint32_t* cnt     = thr + G;
    uint32_t* cand    = cnt + G;

    const int zn = G * NBINS + 4 * G;
    k_zero<<<(zn + BLK - 1) / BLK, BLK, 0, stream>>>(hist, zn);

    const int blocks = (A + BLK - 1) / BLK;
    k_hist<<<blocks, BLK, 0, stream>>>(pd_scores, (const float4*)pd_bboxes, (const float2*)anc,
                                       (const float4*)gt_bboxes, gt_labels, hist, A, G, C);
    k_findbin<<<G, BLK, 0, stream>>>(hist, binIdx, needArr, G);
    k_collect<<<blocks, BLK, 0, stream>>>(pd_scores, (const float4*)pd_bboxes, (const float2*)anc,
                                          (const float4*)gt_bboxes, gt_labels, binIdx, cnt, cand, A, G, C);
    k_thresh<<<1, G_MAX, 0, stream>>>(binIdx, needArr, cnt, cand, thr, G);
    k_final<<<blocks, BLK, 0, stream>>>(pd_scores, (const float4*)pd_bboxes, (const float2*)anc,
                                        (const float4*)gt_bboxes, gt_labels, thr, (float*)d_out, A, G, C);
}